// Model_68513318306322
// MI455X (gfx1250) — compile-verified
//
#include <hip/hip_runtime.h>
#include <stdint.h>

// ---------------------------------------------------------------------------
// Embedding gather  out[t, :] = W[idx[t], :]   (t = 0..B*S-1, D = 256)
// CDNA5 Tensor Data Mover in GATHER mode:
//   - one D# descriptor gathers 16 rows (16-bit row indices) into LDS
//   - row indices are fetched/packed by wave 0's vector lanes (1 load clause,
//     2 VALU packs) and moved to SGPRs with v_readlane — no scalar-load chain
//   - completion tracked with TENSORcnt (s_wait_tensorcnt)
//   - LDS tile streamed out with ds_load_b128 + non-temporal global b128 stores
// ---------------------------------------------------------------------------

typedef __attribute__((ext_vector_type(4))) unsigned int v4u;
typedef __attribute__((ext_vector_type(4))) int          v4i;
typedef __attribute__((ext_vector_type(8))) int          v8i;
typedef __attribute__((ext_vector_type(4))) float        v4f;

#define D_DIM         256                      // row length (floats)
#define VOCAB_ROWS    10001                    // V + 1
#define ROWS_PER_DESC 16                       // gather mode: 16 x 16-bit indices
#define DESCS_PER_WG  2
#define TOKENS_PER_WG (ROWS_PER_DESC * DESCS_PER_WG)   // 32
#define ROW_BYTES     (D_DIM * 4)              // 1024 B
#define TILE_BYTES    (ROWS_PER_DESC * ROW_BYTES)      // 16 KB per descriptor
#define THREADS       256
#define V4_PER_TILE   (TOKENS_PER_WG * D_DIM / 4)      // 2048
#define V4_PER_THREAD (V4_PER_TILE / THREADS)          // 8

__device__ __forceinline__ int imin(int a, int b) { return a < b ? a : b; }

__global__ __launch_bounds__(THREADS) void Model_68513318306322_kernel(
    const int* __restrict__ idx, const float* __restrict__ W,
    float* __restrict__ out, int n_tokens) {
  __shared__ float smem[TOKENS_PER_WG * D_DIM];  // 32 KB (of 320 KB/WGP)

  const int wg_token0 = (int)blockIdx.x * TOKENS_PER_WG;

  // ---- Phase 1: wave 0 builds and issues TDM gather descriptors ------------
  // readfirstlane makes the predicate wave-uniform -> scalar branch, so waves
  // 1..7 jump over this block (TDM ignores EXEC; exec-masking would issue 8x).
  if (__builtin_amdgcn_readfirstlane((int)threadIdx.x) < 32) {
    // Each lane fetches one token PAIR and packs it as {hi16, lo16}.
    // Lanes 0..15 cover the 16 pairs (32 tokens); lanes 16..31 are redundant.
    const int lane = (int)threadIdx.x & 31;
    const int pair = lane & 15;
    const int last = n_tokens - 1;
    const int a    = imin(wg_token0 + 2 * pair,     last);  // v_min, branchless
    const int b    = imin(wg_token0 + 2 * pair + 1, last);
    const unsigned lo     = (unsigned)idx[a];               // global_load_b32
    const unsigned hi     = (unsigned)idx[b];               // global_load_b32
    const int      packed = (int)((lo & 0xFFFFu) | (hi << 16));  // 2 VALU

    const uint64_t wbase    = (uint64_t)(uintptr_t)W;
    // Low 32 bits of the flat shared-segment address == LDS byte offset
    const uint32_t lds_base = (uint32_t)(uintptr_t)&smem[0];
    const uint32_t ga_lo    = (uint32_t)(wbase & 0xFFFFFFFFu);
    const uint32_t ga_hi    = (uint32_t)((wbase >> 32) & 0x01FFFFFFu) |
                              (2u << 30);  // type=2

#pragma unroll
    for (int d = 0; d < DESCS_PER_WG; ++d) {
      const int t0 = wg_token0 + d * ROWS_PER_DESC;
      if (t0 < n_tokens) {
        const int rows = imin(n_tokens - t0, ROWS_PER_DESC);

        // ---- D# groups 2/3: pull 8 packed dwords into SGPRs via readlane ---
        v4i g2 = {__builtin_amdgcn_readlane(packed, 8 * d + 0),
                  __builtin_amdgcn_readlane(packed, 8 * d + 1),
                  __builtin_amdgcn_readlane(packed, 8 * d + 2),
                  __builtin_amdgcn_readlane(packed, 8 * d + 3)};
        v4i g3 = {__builtin_amdgcn_readlane(packed, 8 * d + 4),
                  __builtin_amdgcn_readlane(packed, 8 * d + 5),
                  __builtin_amdgcn_readlane(packed, 8 * d + 6),
                  __builtin_amdgcn_readlane(packed, 8 * d + 7)};

        // ---- D# group 0: count=1 | gather_mode=1 | 16-bit idx; lds; base ---
        v4u g0;
        g0.x = 0x80000001u;
        g0.y = lds_base + (uint32_t)(d * TILE_BYTES);
        g0.z = ga_lo;
        g0.w = ga_hi;

        // ---- D# group 1: sizes / strides -----------------------------------
        v8i g1;
        g1[0] = (int)(2u << 16);                    // wg_mask=0, data_size=4B
        g1[1] = (int)((uint32_t)D_DIM << 16);       // tensor_dim0 = 256
        g1[2] = (int)((uint32_t)VOCAB_ROWS << 16);  // tensor_dim1 = 10001
        g1[3] = (int)((uint32_t)D_DIM << 16);       // tile_dim0   = 256
        g1[4] = rows;                               // tile_dim1 = #valid idx
        g1[5] = D_DIM;                              // tensor_dim0_stride = 256
        g1[6] = 0;
        g1[7] = 0;

#if defined(__clang_major__) && (__clang_major__ >= 23)
        v8i g4 = {0, 0, 0, 0, 0, 0, 0, 0};
        __builtin_amdgcn_tensor_load_to_lds(g0, g1, g2, g3, g4, 0);
#else
        __builtin_amdgcn_tensor_load_to_lds(g0, g1, g2, g3, 0);
#endif
      }
    }
    __builtin_amdgcn_s_wait_tensorcnt(0);
  }
  __syncthreads();

  // ---- Phase 2: stream the 32 KB LDS tile to global ------------------------
  // 2048 float4s / 256 threads = 8 per thread; ds_load_b128 + non-temporal
  // global_store_b128 (output is write-once; keep L2 for the W table).
  const v4f* s4 = (const v4f*)smem;
  v4f*       o4 = (v4f*)(out + (size_t)wg_token0 * D_DIM);

  if (wg_token0 + TOKENS_PER_WG <= n_tokens) {
    // Fast path: full tile, no per-element guards.
#pragma unroll
    for (int i = 0; i < V4_PER_THREAD; ++i) {
      const int e = (int)threadIdx.x + i * THREADS;
      __builtin_nontemporal_store(s4[e], &o4[e]);
    }
  } else {
    // Tail tile: guard per float4.
#pragma unroll
    for (int i = 0; i < V4_PER_THREAD; ++i) {
      const int e     = (int)threadIdx.x + i * THREADS;
      const int token = wg_token0 + e / (D_DIM / 4);
      if (token < n_tokens) {
        __builtin_nontemporal_store(s4[e], &o4[e]);
      }
    }
  }
}

extern "C" void kernel_launch(void* const* d_in, const int* in_sizes, int n_in,
                              void* d_out, int out_size, void* d_ws, size_t ws_size,
                              hipStream_t stream) {
  (void)n_in; (void)out_size; (void)d_ws; (void)ws_size;
  const int*   idx = (const int*)d_in[0];    // input_sentences [B*S]
  const float* W   = (const float*)d_in[1];  // [V+1, 256]
  float*       out = (float*)d_out;          // [B*S, 256]
  const int n_tokens = in_sizes[0];          // 8192

  const int blocks = (n_tokens + TOKENS_PER_WG - 1) / TOKENS_PER_WG;  // 256
  Model_68513318306322_kernel<<<blocks, THREADS, 0, stream>>>(idx, W, out,
                                                              n_tokens);
}